// GCN_78658031059286
// MI455X (gfx1250) — compile-verified
//
#include <hip/hip_runtime.h>
#include <hip/hip_bf16.h>

#define NN 100000
#define EE 1600000
#define DD 128
#define EPSV 1e-5f
#define KCHUNK 64
#define MTILES (NN / 16)   // 6250, exact

typedef __attribute__((ext_vector_type(2))) float v2f;
typedef __attribute__((ext_vector_type(8))) float v8f;

// ---------------- degree / norm ----------------
__global__ void deg_kernel(const int* __restrict__ dst, float* __restrict__ deg) {
    int e = blockIdx.x * blockDim.x + threadIdx.x;
    if (e < EE) atomicAdd(&deg[dst[e]], 1.0f);
}

__global__ void norm_kernel(const float* __restrict__ deg, float* __restrict__ nrm) {
    int n = blockIdx.x * blockDim.x + threadIdx.x;
    if (n < NN) {
        float d = deg[n];
        nrm[n] = rsqrtf(d < 1.0f ? 1.0f : d);
    }
}

// ---------------- propagation: ACC[dst] += norm[src] * X[src] ----------------
// one wave32 per edge; each lane moves a float4 (32 lanes * 4 = 128 = D)
__global__ void scatter_kernel(const float* __restrict__ X, const int* __restrict__ src,
                               const int* __restrict__ dst, const float* __restrict__ nrm,
                               float* __restrict__ acc) {
    int gid  = blockIdx.x * blockDim.x + threadIdx.x;
    int e    = gid >> 5;
    int lane = gid & 31;
    if (e >= EE) return;
    int s = src[e];
    int d = dst[e];
    float ns = nrm[s];
    const float4* xp = (const float4*)(X + (size_t)s * DD);
    float4 v = xp[lane];
    float* op = acc + (size_t)d * DD + lane * 4;
    atomicAdd(op + 0, v.x * ns);
    atomicAdd(op + 1, v.y * ns);
    atomicAdd(op + 2, v.z * ns);
    atomicAdd(op + 3, v.w * ns);
}

__global__ void scale_rows_kernel(float* __restrict__ acc, const float* __restrict__ nrm) {
    int gid = blockIdx.x * blockDim.x + threadIdx.x;  // over N*D
    if (gid < NN * DD) acc[gid] *= nrm[gid >> 7];
}

// ---------------- fused concat-matmul + bias + relu via fp32 WMMA ----------------
// OUT[N,128] = relu([X0|X1|X2][N,384] @ W[384,128] + bias)
// 8 waves/block, each wave computes a 16x128 tile with 8 v_wmma_f32_16x16x4_f32 accs.
// W is staged into LDS K-pair-interleaved + bank-swizzled so each B fragment is a
// single conflict-free ds_load_b64 into an even-aligned VGPR pair.
__global__ __launch_bounds__(256) void tag_matmul_kernel(
    const float* __restrict__ X0, const float* __restrict__ X1, const float* __restrict__ X2,
    const float* __restrict__ W, const float* __restrict__ bias, float* __restrict__ out) {
    // lw2[p][c'] = { W[2p][c], W[2p+1][c] } with c' = c ^ ((p&1)<<4)  (32 KB)
    __shared__ v2f lw2[(KCHUNK / 2) * DD];

    int tid   = threadIdx.x;
    int wave  = tid >> 5;
    int lane  = tid & 31;
    int tileM = blockIdx.x * 8 + wave;
    bool valid = tileM < MTILES;

    int row16 = lane & 15;      // M (for A) / N (for B,C,D) within tile
    int half  = lane >> 4;      // 0: K 0..1 / M 0..7 ; 1: K 2..3 / M 8..15
    int kb    = half * 2;

    v8f acc[8] = {};

    #pragma unroll 1
    for (int seg = 0; seg < 3; ++seg) {
        // phi-select of global pointers (keeps addrspace(1) -> global_load, not flat_load)
        const float* X = X0;
        if (seg == 1) X = X1;
        if (seg == 2) X = X2;
        const float* xrow = X + (size_t)(tileM * 16 + row16) * DD;

        #pragma unroll 1
        for (int kc = 0; kc < DD; kc += KCHUNK) {
            // stage W[(seg*128+kc)..+64][0..128] into LDS, K-pair interleaved
            for (int i = tid; i < (KCHUNK / 2) * DD; i += 256) {
                int p = i >> 7;        // K pair row 0..31
                int c = i & (DD - 1);  // column
                const float* wp = W + (size_t)(seg * DD + kc + 2 * p) * DD + c;
                v2f wv;
                wv.x = wp[0];
                wv.y = wp[DD];
                lw2[(p << 7) + (c ^ ((p & 1) << 4))] = wv;
            }
            __syncthreads();

            if (valid) {
                #pragma unroll 4
                for (int kk = 0; kk < KCHUNK; kk += 4) {
                    // A fragment: 16x4, lane holds A[m][kb], A[m][kb+1]
                    v2f a;
                    a.x = xrow[kc + kk + kb];
                    a.y = xrow[kc + kk + kb + 1];
                    // B fragments: pair row (kk+kb)/2, swizzled column tile t^half
                    int prow = (kk >> 1) + half;
                    const v2f* bp = lw2 + (prow << 7) + row16;
                    #pragma unroll
                    for (int t = 0; t < 8; ++t) {
                        v2f b = bp[(t ^ half) * 16];
                        acc[t] = __builtin_amdgcn_wmma_f32_16x16x4_f32(
                            false, a, false, b, (short)0, acc[t], false, false);
                    }
                }
            }
            __syncthreads();
        }
    }

    if (valid) {
        #pragma unroll
        for (int t = 0; t < 8; ++t) {
            #pragma unroll
            for (int j = 0; j < 8; ++j) {
                int m = tileM * 16 + half * 8 + j;   // C/D layout: VGPR j -> M=j (+8 for upper half)
                int n = t * 16 + row16;
                float v = acc[t][j] + bias[n];
                out[(size_t)m * DD + n] = v > 0.0f ? v : 0.0f;
            }
        }
    }
}

// ---------------- batchnorm (training-mode batch stats) ----------------
__global__ void bn_stats_kernel(const float* __restrict__ F, float* __restrict__ ssum,
                                float* __restrict__ ssq) {
    int c = threadIdx.x;  // 128 threads, one per column
    float s = 0.f, q = 0.f;
    for (int r = blockIdx.x; r < NN; r += gridDim.x) {
        float v = F[(size_t)r * DD + c];
        s += v;
        q += v * v;
    }
    atomicAdd(&ssum[c], s);
    atomicAdd(&ssq[c], q);
}

__global__ void bn_finalize_kernel(const float* __restrict__ ssum, const float* __restrict__ ssq,
                                   float* __restrict__ mean, float* __restrict__ inv) {
    int c = threadIdx.x;
    if (c < DD) {
        float m = ssum[c] / (float)NN;
        float v = ssq[c] / (float)NN - m * m;
        mean[c] = m;
        inv[c] = rsqrtf(v + EPSV);
    }
}

__global__ void bn_apply_kernel(float* __restrict__ F, const float* __restrict__ mean,
                                const float* __restrict__ inv) {
    int gid = blockIdx.x * blockDim.x + threadIdx.x;
    if (gid < NN * DD) {
        int c = gid & (DD - 1);
        F[gid] = (F[gid] - mean[c]) * inv[c];
    }
}

// ---------------- output head: [X0|X1|X2] @ Wout[384,3] + bout, softmax ----------------
__global__ void out_kernel(const float* __restrict__ X0, const float* __restrict__ X1,
                           const float* __restrict__ X2, const float* __restrict__ Wout,
                           const float* __restrict__ bout, float* __restrict__ out) {
    int gid  = blockIdx.x * blockDim.x + threadIdx.x;
    int node = gid >> 5;
    int lane = gid & 31;
    if (node >= NN) return;
    const float* r0 = X0 + (size_t)node * DD;
    const float* r1 = X1 + (size_t)node * DD;
    const float* r2 = X2 + (size_t)node * DD;
    float p0 = 0.f, p1 = 0.f, p2 = 0.f;
    for (int k = lane; k < DD; k += 32) {
        float x0 = r0[k], x1 = r1[k], x2 = r2[k];
        const float* w0 = Wout + (size_t)k * 3;
        const float* w1 = Wout + (size_t)(k + DD) * 3;
        const float* w2 = Wout + (size_t)(k + 2 * DD) * 3;
        p0 += x0 * w0[0] + x1 * w1[0] + x2 * w2[0];
        p1 += x0 * w0[1] + x1 * w1[1] + x2 * w2[1];
        p2 += x0 * w0[2] + x1 * w1[2] + x2 * w2[2];
    }
    #pragma unroll
    for (int off = 16; off >= 1; off >>= 1) {
        p0 += __shfl_xor(p0, off, 32);
        p1 += __shfl_xor(p1, off, 32);
        p2 += __shfl_xor(p2, off, 32);
    }
    if (lane == 0) {
        p0 += bout[0]; p1 += bout[1]; p2 += bout[2];
        float m = fmaxf(p0, fmaxf(p1, p2));
        float e0 = __expf(p0 - m), e1 = __expf(p1 - m), e2 = __expf(p2 - m);
        float s = 1.0f / (e0 + e1 + e2);
        out[(size_t)node * 3 + 0] = e0 * s;
        out[(size_t)node * 3 + 1] = e1 * s;
        out[(size_t)node * 3 + 2] = e2 * s;
    }
}

extern "C" void kernel_launch(void* const* d_in, const int* in_sizes, int n_in,
                              void* d_out, int out_size, void* d_ws, size_t ws_size,
                              hipStream_t stream) {
    const float* in_feat = (const float*)d_in[0];
    const int*   src     = (const int*)d_in[1];
    const int*   dst     = (const int*)d_in[2];
    const float* W1 = (const float*)d_in[3];
    const float* b1 = (const float*)d_in[4];
    const float* W2 = (const float*)d_in[5];
    const float* b2 = (const float*)d_in[6];
    const float* W3 = (const float*)d_in[7];
    const float* b3 = (const float*)d_in[8];
    const float* Wout = (const float*)d_in[9];
    const float* bout = (const float*)d_in[10];

    char* ws = (char*)d_ws;
    const size_t NB = (size_t)NN * DD * sizeof(float);  // 51.2 MB
    float* F0   = (float*)(ws + 0 * NB);
    float* F1   = (float*)(ws + 1 * NB);
    float* A1   = (float*)(ws + 2 * NB);
    float* A2   = (float*)(ws + 3 * NB);
    float* nrm  = (float*)(ws + 4 * NB);
    float* deg  = nrm + NN;
    float* ssum = deg + NN;
    float* ssq  = ssum + DD;
    float* smean = ssq + DD;
    float* sinv  = smean + DD;

    const int threads = 256;
    const int edgeWaveBlocks = (int)(((long)EE * 32 + threads - 1) / threads);
    const int elemBlocks = (NN * DD + threads - 1) / threads;
    const int mmBlocks = (MTILES + 7) / 8;

    // degree + symmetric norm
    hipMemsetAsync(deg, 0, (size_t)NN * sizeof(float), stream);
    deg_kernel<<<(EE + threads - 1) / threads, threads, 0, stream>>>(dst, deg);
    norm_kernel<<<(NN + threads - 1) / threads, threads, 0, stream>>>(deg, nrm);

    auto prop = [&](const float* X, float* ACC) {
        hipMemsetAsync(ACC, 0, NB, stream);
        scatter_kernel<<<edgeWaveBlocks, threads, 0, stream>>>(X, src, dst, nrm, ACC);
        scale_rows_kernel<<<elemBlocks, threads, 0, stream>>>(ACC, nrm);
    };

    // layer 1: in_feat -> F0
    prop(in_feat, A1);
    prop(A1, A2);
    tag_matmul_kernel<<<mmBlocks, threads, 0, stream>>>(in_feat, A1, A2, W1, b1, F0);

    // layer 2: F0 -> F1
    prop(F0, A1);
    prop(A1, A2);
    tag_matmul_kernel<<<mmBlocks, threads, 0, stream>>>(F0, A1, A2, W2, b2, F1);

    // layer 3: F1 -> F0
    prop(F1, A1);
    prop(A1, A2);
    tag_matmul_kernel<<<mmBlocks, threads, 0, stream>>>(F1, A1, A2, W3, b3, F0);

    // batchnorm (in place on F0)
    hipMemsetAsync(ssum, 0, 2 * DD * sizeof(float), stream);
    bn_stats_kernel<<<1024, DD, 0, stream>>>(F0, ssum, ssq);
    bn_finalize_kernel<<<1, DD, 0, stream>>>(ssum, ssq, smean, sinv);
    bn_apply_kernel<<<elemBlocks, threads, 0, stream>>>(F0, smean, sinv);

    // output head + softmax
    prop(F0, A1);
    prop(A1, A2);
    const int nodeWaveBlocks = (int)(((long)NN * 32 + threads - 1) / threads);
    out_kernel<<<nodeWaveBlocks, threads, 0, stream>>>(F0, A1, A2, Wout, bout, (float*)d_out);
}